// EnhancementEncoder_36043365548797
// MI455X (gfx1250) — compile-verified
//
#include <hip/hip_runtime.h>
#include <cstdint>
#include <cstddef>

// Problem constants (match reference shapes)
#define B_ 2
#define D_ 32
#define C_ 64
#define H_ 72
#define W_ 136
constexpr int HW    = H_ * W_;          // 9792
constexpr int NPIX  = B_ * HW;          // 19584
constexpr int PIXPB = 8;                // pixels per block (one wave32 each)
constexpr int THREADS = PIXPB * 32;     // 256 threads = 8 waves

#if defined(__has_builtin)
#if __has_builtin(__builtin_amdgcn_tensor_load_to_lds)
#define USE_TDM 1
#endif
#endif
#ifndef USE_TDM
#define USE_TDM 0
#endif

typedef unsigned int u32x4 __attribute__((ext_vector_type(4)));
typedef int          i32x4 __attribute__((ext_vector_type(4)));
typedef int          i32x8 __attribute__((ext_vector_type(8)));
typedef float        vf2   __attribute__((ext_vector_type(2)));

// LDS layout of the staged current-feature tile:
//  - TDM path writes the 2D tile row-major (pixel fastest): scur[c*8 + p]
//  - async fallback writes per-thread:                      scur[p*64 + c]
#if USE_TDM
#define SC_IDX(p, c) ((c) * PIXPB + (p))
#else
#define SC_IDX(p, c) ((p) * C_ + (c))
#endif

__global__ __launch_bounds__(THREADS)
void cost_volume_fisheye_kernel(
    const float* __restrict__ cur,    // [B,C,H,W]
    const float* __restrict__ lf,     // [B,1,C,H,W]
    const float* __restrict__ prior,  // [B,1,H,W]
    const float* __restrict__ sfac,   // [H,W]
    const float* __restrict__ theta,  // [B,H,W]
    const float* __restrict__ angle,  // [B,H,W]
    const float* __restrict__ poses,  // [B,1,4,4]
    const float* __restrict__ intr,   // [B,3,3]
    const float* __restrict__ dist,   // [B,4]
    float* __restrict__ out)          // [2, B, D, H, W] (cost_volume ++ missing)
{
    __shared__ float scur[PIXPB * C_];          // staged current_feats, 2 KB
    const int pix0 = blockIdx.x * PIXPB;

#if USE_TDM
    // ---- Stage current_feats via the Tensor Data Mover: one 2D tile
    //      (tile_dim0 = 8 contiguous pixels, tile_dim1 = 64 channels,
    //       dim0 line stride = H*W) issued by wave 0 only (TENSORcnt path).
    if ((threadIdx.x >> 5) == 0) {
        const int b0   = pix0 / HW;
        const int rem0 = pix0 - b0 * HW;
        const unsigned long long ga =
            (unsigned long long)(const void*)(cur + (size_t)b0 * C_ * HW + rem0);
        const unsigned lds_addr = (unsigned)(size_t)(&scur[0]);

        // D# group 0: count=1 | lds_addr | global_addr(57b) | type=2
        u32x4 g0 = { 1u,
                     lds_addr,
                     (unsigned)(ga & 0xFFFFFFFFull),
                     (unsigned)((ga >> 32) & 0x01FFFFFFull) | 0x80000000u };
        // D# group 1: data_size=4B(2<<16); tensor_dim0=HW; tensor_dim1=C;
        //             tile_dim0=8; tile_dim1=64; tensor_dim0_stride=HW
        i32x8 g1 = { (int)(2u << 16),                       // wg_mask=0, data_size=2
                     (int)(((unsigned)HW & 0xFFFFu) << 16), // tensor_dim0 lo16 << 16
                     (int)(((unsigned)C_ & 0xFFFFu) << 16)  // tensor_dim1 lo16 << 16
                         | (int)((unsigned)HW >> 16),       //  | tensor_dim0 hi16
                     (int)((unsigned)PIXPB << 16),          // tile_dim0 << 16
                     (int)C_,                               // tile_dim1 (tile_dim2=0)
                     (int)HW,                               // tensor_dim0_stride lo32
                     0, 0 };
        i32x4 g2 = { 0, 0, 0, 0 };                          // 2D tile: dims 2+ unused
        i32x4 g3 = { 0, 0, 0, 0 };
        i32x8 g4 = { 0, 0, 0, 0, 0, 0, 0, 0 };              // unused trailing group
        __builtin_amdgcn_tensor_load_to_lds(g0, g1, g2, g3, g4, 0);
    }
#if __has_builtin(__builtin_amdgcn_s_wait_tensorcnt)
    __builtin_amdgcn_s_wait_tensorcnt(0);
#else
    asm volatile("s_wait_tensorcnt 0" ::: "memory");
#endif
#else
    // ---- Fallback: per-lane async global->LDS DMA (ASYNCcnt path).
    #pragma unroll
    for (int j = 0; j < (PIXPB * C_) / THREADS; ++j) {
        int idx = threadIdx.x + j * THREADS;    // 0..511
        int p   = idx >> 6;                     // pixel slot 0..7
        int c   = idx & (C_ - 1);               // channel
        int pix = pix0 + p;
        if (pix >= NPIX) pix = NPIX - 1;
        int b   = pix / HW;
        int rem = pix - b * HW;                 // h*W + w
        unsigned lds_addr = (unsigned)(size_t)(&scur[SC_IDX(p, c)]);
        unsigned long long gaddr =
            (unsigned long long)(const void*)(cur + (size_t)(b * C_ + c) * HW + rem);
        asm volatile("global_load_async_to_lds_b32 %0, %1, off"
                     :: "v"(lds_addr), "v"(gaddr) : "memory");
    }
#if __has_builtin(__builtin_amdgcn_s_wait_asynccnt)
    __builtin_amdgcn_s_wait_asynccnt(0);
#else
    asm volatile("s_wait_asynccnt 0" ::: "memory");
#endif
#endif
    __syncthreads();

    // ---- One wave per pixel; lane index == depth hypothesis d (D == wave32).
    const int wave = threadIdx.x >> 5;
    const int d    = threadIdx.x & 31;
    const int pix  = pix0 + wave;
    if (pix >= NPIX) return;
    const int b   = pix / HW;
    const int rem = pix - b * HW;
    const int h   = rem / W_;
    const int w   = rem - h * W_;

    // Wave-uniform per-pixel scalars (SGPR-resident).
    const float pd  = prior[b * HW + rem];
    const float sf  = sfac[rem];
    const float thl = theta[b * HW + rem];
    const float anl = angle[b * HW + rem];

    const float smin = pd / (1.0f + sf);
    const float smax = pd * (1.0f + sf);
    const float inv_smax = 1.0f / smax;
    const float inv_smin = 1.0f / smin;

    // Per-lane warp depth: inverse-depth interpolation, itv = d/(D-1)
    const float itv = (float)d * (1.0f / (float)(D_ - 1));
    const float wd  = 1.0f / (inv_smax + (inv_smin - inv_smax) * itv);

    const float st = sinf(thl), ct = cosf(thl);
    const float ca = cosf(anl), sa = sinf(anl);
    const float dirx = st * ca, diry = st * sa, dirz = ct;

    const float* P = poses + b * 16;
    const float cxm = wd * dirx, cym = wd * diry, czm = wd * dirz;
    const float X = P[0]*cxm + P[1]*cym + P[2] *czm + P[3];
    const float Y = P[4]*cxm + P[5]*cym + P[6] *czm + P[7];
    const float Z = P[8]*cxm + P[9]*cym + P[10]*czm + P[11];

    const float r   = sqrtf(X * X + Y * Y) + 1e-8f;
    const float thp = atan2f(r, Z);
    const float t2  = thp * thp;
    const float* K  = dist + b * 4;
    const float rdv = thp * (1.0f + t2 * (K[0] + t2 * (K[1] + t2 * (K[2] + t2 * K[3]))));
    const float* I  = intr + b * 9;
    const float u = I[0] * rdv * X / r + I[2];
    const float v = I[4] * rdv * Y / r + I[5];

    // Edge mask (stricter than bilinear validity): inside it, all 4 corners are
    // in-bounds, so the hot loop needs no clamping or per-corner valid weights.
    const bool edge = (u >= 2.0f) && (u <= (float)(W_ - 2)) &&
                      (v >= 2.0f) && (v <= (float)(H_ - 2)) &&
                      (h >= 2) && (h <= H_ - 3) && (w >= 2) && (w <= W_ - 3);

    float diffs = 0.0f;
    if (edge) {
        const float x0f = floorf(u), y0f = floorf(v);
        const float wx1 = u - x0f,   wy1 = v - y0f;
        const float wx0 = 1.0f - wx1, wy0 = 1.0f - wy1;
        const int   x0  = (int)x0f,  y0  = (int)y0f;
        const float w00 = wx0 * wy0, w01 = wx1 * wy0;
        const float w10 = wx0 * wy1, w11 = wx1 * wy1;

        const float* __restrict__ f00 = lf + (size_t)b * (C_ * HW) + (y0 * W_ + x0);
        __builtin_prefetch(f00, 0, 0);          // global_prefetch_b8
        __builtin_prefetch(f00 + W_, 0, 0);

        float acc = 0.0f;
        #pragma unroll 8
        for (int c = 0; c < C_; ++c) {
            const float* f = f00 + c * HW;
            vf2 top, bot;                        // adjacent corner pairs ->
            __builtin_memcpy(&top, f,      8);   //   one b64 gather per row
            __builtin_memcpy(&bot, f + W_, 8);
            const float s = w00 * top.x + w01 * top.y
                          + w10 * bot.x + w11 * bot.y;
            acc += fabsf(s - scur[SC_IDX(wave, c)]);
        }
        diffs = acc * (1.0f / (float)C_);
    }

    // Epilogue: counts/missing + max-over-D via wave32 shuffle reduction.
    const float counts  = (diffs > 0.0f) ? 1.0f : 0.0f;
    const float cost    = diffs / (counts + 1e-7f);
    const float missing = (cost == 0.0f) ? 1.0f : 0.0f;

    float m = cost;
    #pragma unroll
    for (int off = 16; off >= 1; off >>= 1)
        m = fmaxf(m, __shfl_xor(m, off, 32));

    const float final_cost = (missing != 0.0f) ? m : cost;

    const size_t oidx = ((size_t)(b * D_ + d) * H_ + h) * W_ + w;
    out[oidx] = final_cost;
    out[(size_t)B_ * D_ * HW + oidx] = missing;
}

extern "C" void kernel_launch(void* const* d_in, const int* in_sizes, int n_in,
                              void* d_out, int out_size, void* d_ws, size_t ws_size,
                              hipStream_t stream) {
    (void)in_sizes; (void)n_in; (void)out_size; (void)d_ws; (void)ws_size;
    const float* cur   = (const float*)d_in[0];  // current_feats
    const float* lf    = (const float*)d_in[1];  // lookup_feats
    const float* prior = (const float*)d_in[2];  // prior_depth
    const float* sfac  = (const float*)d_in[3];  // scale_facs
    const float* theta = (const float*)d_in[4];  // theta_lut
    const float* angle = (const float*)d_in[5];  // angle_lut
    const float* poses = (const float*)d_in[6];  // lookup_poses
    const float* intr  = (const float*)d_in[7];  // intrinsics
    const float* dist  = (const float*)d_in[8];  // distortion_coeffs
    float* out = (float*)d_out;

    const int blocks = (NPIX + PIXPB - 1) / PIXPB;   // 2448
    hipLaunchKernelGGL(cost_volume_fisheye_kernel, dim3(blocks), dim3(THREADS), 0, stream,
                       cur, lf, prior, sfac, theta, angle, poses, intr, dist, out);
}